// GAT_36000415875142
// MI455X (gfx1250) — compile-verified
//
#include <hip/hip_runtime.h>
#include <math.h>

#define NN   50000
#define EE   400000
#define ET   450000    // edges + self loops
#define INC  128
#define HH   4
#define CC   32
#define HIDD 128
#define CLS  40
#define CLSP 48
#define EPSV 1e-5f
#define SLOPE 0.2f

typedef __attribute__((ext_vector_type(2))) float v2f;
typedef __attribute__((ext_vector_type(8))) float v8f;

// ---------------- utility ----------------
__global__ void fill_kernel(float* __restrict__ p, float v, long long n) {
  long long i = (long long)blockIdx.x * blockDim.x + threadIdx.x;
  if (i < n) p[i] = v;
}

__device__ __forceinline__ void atomicMaxF(float* addr, float v) {
  if (v >= 0.f) atomicMax((int*)addr, __float_as_int(v));
  else          atomicMin((unsigned int*)addr, __float_as_uint(v));
}

__device__ __forceinline__ float lrelu(float e) { return e > 0.f ? e : SLOPE * e; }

// ---------------- WMMA fp32 GEMM, LDS-staged B (K-paired layout) ----------------
// One wave computes a 16-row x OC output stripe (OC/16 accumulator tiles).
// Block = 160 threads (5 waves, 5 stripes); B[K x OC] staged in LDS as
// Bs[(k/2)][n][2] so each WMMA B fragment is a single aligned ds_load_b64.
template<int K, int OC>
__global__ void gemm_wmma_blk(const float* __restrict__ A, const float* __restrict__ B,
                              float* __restrict__ C) {
  constexpr int NT = OC / 16;
  __shared__ float Bs[K * OC];
  for (int i = threadIdx.x; i < K * OC; i += blockDim.x) {
    int k = i / OC, n = i - k * OC;
    Bs[(k >> 1) * (2 * OC) + n * 2 + (k & 1)] = B[i];
  }
  __syncthreads();

  const int wave = threadIdx.x >> 5;
  const int lane = threadIdx.x & 31;
  const int half = lane >> 4;       // K sub-pair select
  const int l15  = lane & 15;
  const int row0 = (blockIdx.x * 5 + wave) << 4;

  v8f acc[NT];
#pragma unroll
  for (int t = 0; t < NT; ++t) acc[t] = (v8f){};

  const float* Ap = A + (size_t)(row0 + l15) * K + (half << 1);
  // pair row p = k0/2 + half ; element (col)*2
  const float* Bp = Bs + half * (2 * OC) + l15 * 2;
#pragma unroll 4
  for (int k0 = 0; k0 < K; k0 += 4) {
    v2f a;
    a.x = Ap[0]; a.y = Ap[1];
#pragma unroll
    for (int t = 0; t < NT; ++t) {
      v2f b = *(const v2f*)(Bp + t * 32);   // (B[k][n], B[k+1][n]) contiguous
      acc[t] = __builtin_amdgcn_wmma_f32_16x16x4_f32(false, a, false, b,
                                                     (short)0, acc[t], false, false);
    }
    Ap += 4;
    Bp += 2 * (2 * OC);                     // advance 2 pair-rows (4 K values)
  }
#pragma unroll
  for (int t = 0; t < NT; ++t)
#pragma unroll
    for (int j = 0; j < 8; ++j)
      C[(size_t)(row0 + (half << 3) + j) * OC + t * 16 + l15] = acc[t][j];
}

// ---------------- attention scores per node ----------------
template<int H, int C, int XWLD>
__global__ void scores_k(const float* __restrict__ xw, const float* __restrict__ as,
                         const float* __restrict__ ad, float* __restrict__ esrc,
                         float* __restrict__ edst) {
  long long tid = (long long)blockIdx.x * blockDim.x + threadIdx.x;
  if (tid >= (long long)NN * H) return;
  int h = (int)(tid % H);
  long long n = tid / H;
  const float* xp = xw + (size_t)n * XWLD + h * C;
  float ss = 0.f, sd = 0.f;
#pragma unroll 8
  for (int c = 0; c < C; ++c) {
    float v = xp[c];
    ss += v * as[h * C + c];
    sd += v * ad[h * C + c];
  }
  esrc[tid] = ss;
  edst[tid] = sd;
}

// ---------------- H=4 vectorized edge passes ----------------
__global__ void edge_max4_k(const int* __restrict__ src, const int* __restrict__ dst,
                            const float4* __restrict__ esrc, const float4* __restrict__ edst,
                            float4* __restrict__ ebuf, float* __restrict__ mseg) {
  long long eid = (long long)blockIdx.x * blockDim.x + threadIdx.x;
  if (eid >= ET) return;
  int s, d;
  if (eid < EE) { s = src[eid]; d = dst[eid]; } else { s = d = (int)(eid - EE); }
  float4 es = esrc[s], ed = edst[d], e;
  e.x = lrelu(es.x + ed.x);
  e.y = lrelu(es.y + ed.y);
  e.z = lrelu(es.z + ed.z);
  e.w = lrelu(es.w + ed.w);
  ebuf[eid] = e;
  float* mp = mseg + (size_t)d * 4;
  atomicMaxF(mp + 0, e.x);
  atomicMaxF(mp + 1, e.y);
  atomicMaxF(mp + 2, e.z);
  atomicMaxF(mp + 3, e.w);
}

__global__ void edge_exp4_k(const int* __restrict__ dst, float4* __restrict__ ebuf,
                            const float4* __restrict__ mseg, float* __restrict__ den) {
  long long eid = (long long)blockIdx.x * blockDim.x + threadIdx.x;
  if (eid >= ET) return;
  int d = (eid < EE) ? dst[eid] : (int)(eid - EE);
  float4 e = ebuf[eid], mx = mseg[d], ex;
  ex.x = __expf(e.x - mx.x);
  ex.y = __expf(e.y - mx.y);
  ex.z = __expf(e.z - mx.z);
  ex.w = __expf(e.w - mx.w);
  ebuf[eid] = ex;
  float* dp = den + (size_t)d * 4;
  atomicAdd(dp + 0, ex.x);
  atomicAdd(dp + 1, ex.y);
  atomicAdd(dp + 2, ex.z);
  atomicAdd(dp + 3, ex.w);
}

// ---------------- H=1 scalar edge passes (layer 3) ----------------
__global__ void edge_max1_k(const int* __restrict__ src, const int* __restrict__ dst,
                            const float* __restrict__ esrc, const float* __restrict__ edst,
                            float* __restrict__ ebuf, float* __restrict__ mseg) {
  long long eid = (long long)blockIdx.x * blockDim.x + threadIdx.x;
  if (eid >= ET) return;
  int s, d;
  if (eid < EE) { s = src[eid]; d = dst[eid]; } else { s = d = (int)(eid - EE); }
  float e = lrelu(esrc[s] + edst[d]);
  ebuf[eid] = e;
  atomicMaxF(&mseg[d], e);
}

__global__ void edge_exp1_k(const int* __restrict__ dst, float* __restrict__ ebuf,
                            const float* __restrict__ mseg, float* __restrict__ den) {
  long long eid = (long long)blockIdx.x * blockDim.x + threadIdx.x;
  if (eid >= ET) return;
  int d = (eid < EE) ? dst[eid] : (int)(eid - EE);
  float ex = __expf(ebuf[eid] - mseg[d]);
  ebuf[eid] = ex;
  atomicAdd(&den[d], ex);
}

// ---------------- edge pass 3: alpha * xw[src] -> agg[dst], 4 channels/thread ----------------
template<int H, int C, int XWLD, int OCH>
__global__ void edge_agg4_k(const int* __restrict__ src, const int* __restrict__ dst,
                            const float* __restrict__ ebuf, const float* __restrict__ den,
                            const float* __restrict__ xw, float* __restrict__ agg) {
  constexpr int G = OCH / 4;
  long long tid = (long long)blockIdx.x * blockDim.x + threadIdx.x;
  if (tid >= (long long)ET * G) return;
  int g = (int)(tid % G);
  long long eid = tid / G;
  int c = g * 4;
  int h = c / C;
  int s, d;
  if (eid < EE) { s = src[eid]; d = dst[eid]; } else { s = d = (int)(eid - EE); }
  float alpha = ebuf[(size_t)eid * H + h] / den[(size_t)d * H + h];
  float4 xv = *(const float4*)(xw + (size_t)s * XWLD + c);
  float* ap = agg + (size_t)d * OCH + c;
  atomicAdd(ap + 0, alpha * xv.x);
  atomicAdd(ap + 1, alpha * xv.y);
  atomicAdd(ap + 2, alpha * xv.z);
  atomicAdd(ap + 3, alpha * xv.w);
}

// ---------------- BN + residual + ELU epilogue ----------------
__global__ void bn_res_elu_k(const float* __restrict__ agg, const float* __restrict__ bias,
                             const float* __restrict__ g, const float* __restrict__ be,
                             const float* __restrict__ m, const float* __restrict__ v,
                             const float* __restrict__ res, float* __restrict__ out) {
  long long i = (long long)blockIdx.x * blockDim.x + threadIdx.x;
  if (i >= (long long)NN * HIDD) return;
  int f = (int)(i & (HIDD - 1));
  float h = agg[i] + bias[f];
  h = (h - m[f]) * rsqrtf(v[f] + EPSV) * g[f] + be[f] + res[i];
  out[i] = h > 0.f ? h : (__expf(h) - 1.f);
}

// ---------------- final bias add ----------------
__global__ void out_bias_k(const float* __restrict__ agg, const float* __restrict__ b,
                           float* __restrict__ out) {
  long long i = (long long)blockIdx.x * blockDim.x + threadIdx.x;
  if (i >= (long long)NN * CLS) return;
  int c = (int)(i % CLS);
  out[i] = agg[i] + b[c];
}

// ---------------- pad W3 [128 x 40] -> [128 x 48] ----------------
__global__ void pad_w3_k(const float* __restrict__ W3, float* __restrict__ W3p) {
  int i = blockIdx.x * blockDim.x + threadIdx.x;
  if (i >= HIDD * CLSP) return;
  int k = i / CLSP, c = i % CLSP;
  W3p[i] = (c < CLS) ? W3[k * CLS + c] : 0.f;
}

static inline int blks(long long n, int b) { return (int)((n + b - 1) / b); }

extern "C" void kernel_launch(void* const* d_in, const int* in_sizes, int n_in,
                              void* d_out, int out_size, void* d_ws, size_t ws_size,
                              hipStream_t stream) {
  (void)in_sizes; (void)n_in; (void)out_size; (void)ws_size;
  const float* x    = (const float*)d_in[0];
  const int*   ei   = (const int*)  d_in[1];
  const float* W1   = (const float*)d_in[2];
  const float* as1  = (const float*)d_in[3];
  const float* ad1  = (const float*)d_in[4];
  const float* b1   = (const float*)d_in[5];
  const float* W2   = (const float*)d_in[6];
  const float* as2  = (const float*)d_in[7];
  const float* ad2  = (const float*)d_in[8];
  const float* b2   = (const float*)d_in[9];
  const float* W3   = (const float*)d_in[10];
  const float* as3  = (const float*)d_in[11];
  const float* ad3  = (const float*)d_in[12];
  const float* b3   = (const float*)d_in[13];
  const float* Wres = (const float*)d_in[14];
  const float* g1   = (const float*)d_in[15];
  const float* be1  = (const float*)d_in[16];
  const float* m1   = (const float*)d_in[17];
  const float* v1   = (const float*)d_in[18];
  const float* g2   = (const float*)d_in[19];
  const float* be2  = (const float*)d_in[20];
  const float* m2   = (const float*)d_in[21];
  const float* v2   = (const float*)d_in[22];
  float* out = (float*)d_out;

  const int* src = ei;
  const int* dst = ei + EE;

  // workspace layout (floats)
  float* ws   = (float*)d_ws;
  float* xw   = ws;                         // N*128 (layer3 uses N*48 of it)
  float* agg  = xw   + (size_t)NN * HIDD;   // N*128
  float* hbuf = agg  + (size_t)NN * HIDD;   // N*128  (layer1 out)
  float* h2   = hbuf + (size_t)NN * HIDD;   // N*128  (xres for L1, layer2 out)
  float* ebuf = h2   + (size_t)NN * HIDD;   // ET*4
  float* esrc = ebuf + (size_t)ET * HH;     // N*4
  float* edst = esrc + (size_t)NN * HH;     // N*4
  float* mseg = edst + (size_t)NN * HH;     // N*4
  float* den  = mseg + (size_t)NN * HH;     // N*4
  float* W3p  = den  + (size_t)NN * HH;     // 128*48

  const int TB = 256;
  const int GEMM_BLOCKS = (NN / 16) / 5;    // 3125 stripes / 5 waves = 625
  dim3 gemm_blk(160);

  // ===== layer 1 =====
  gemm_wmma_blk<INC, HIDD><<<GEMM_BLOCKS, gemm_blk, 0, stream>>>(x, W1, xw);
  gemm_wmma_blk<INC, HIDD><<<GEMM_BLOCKS, gemm_blk, 0, stream>>>(x, Wres, h2);
  scores_k<HH, CC, HIDD><<<blks((long long)NN * HH, TB), TB, 0, stream>>>(xw, as1, ad1, esrc, edst);
  fill_kernel<<<blks((long long)NN * HH, TB), TB, 0, stream>>>(mseg, -INFINITY, (long long)NN * HH);
  fill_kernel<<<blks((long long)NN * HH, TB), TB, 0, stream>>>(den, 0.f, (long long)NN * HH);
  fill_kernel<<<blks((long long)NN * HIDD, TB), TB, 0, stream>>>(agg, 0.f, (long long)NN * HIDD);
  edge_max4_k<<<blks(ET, TB), TB, 0, stream>>>(src, dst, (const float4*)esrc, (const float4*)edst,
                                               (float4*)ebuf, mseg);
  edge_exp4_k<<<blks(ET, TB), TB, 0, stream>>>(dst, (float4*)ebuf, (const float4*)mseg, den);
  edge_agg4_k<HH, CC, HIDD, HIDD><<<blks((long long)ET * (HIDD / 4), TB), TB, 0, stream>>>(
      src, dst, ebuf, den, xw, agg);
  bn_res_elu_k<<<blks((long long)NN * HIDD, TB), TB, 0, stream>>>(agg, b1, g1, be1, m1, v1, h2, hbuf);

  // ===== layer 2 =====
  gemm_wmma_blk<HIDD, HIDD><<<GEMM_BLOCKS, gemm_blk, 0, stream>>>(hbuf, W2, xw);
  scores_k<HH, CC, HIDD><<<blks((long long)NN * HH, TB), TB, 0, stream>>>(xw, as2, ad2, esrc, edst);
  fill_kernel<<<blks((long long)NN * HH, TB), TB, 0, stream>>>(mseg, -INFINITY, (long long)NN * HH);
  fill_kernel<<<blks((long long)NN * HH, TB), TB, 0, stream>>>(den, 0.f, (long long)NN * HH);
  fill_kernel<<<blks((long long)NN * HIDD, TB), TB, 0, stream>>>(agg, 0.f, (long long)NN * HIDD);
  edge_max4_k<<<blks(ET, TB), TB, 0, stream>>>(src, dst, (const float4*)esrc, (const float4*)edst,
                                               (float4*)ebuf, mseg);
  edge_exp4_k<<<blks(ET, TB), TB, 0, stream>>>(dst, (float4*)ebuf, (const float4*)mseg, den);
  edge_agg4_k<HH, CC, HIDD, HIDD><<<blks((long long)ET * (HIDD / 4), TB), TB, 0, stream>>>(
      src, dst, ebuf, den, xw, agg);
  bn_res_elu_k<<<blks((long long)NN * HIDD, TB), TB, 0, stream>>>(agg, b2, g2, be2, m2, v2, hbuf, h2);

  // ===== layer 3 (1 head, CLS=40, padded GEMM to 48 cols) =====
  pad_w3_k<<<blks(HIDD * CLSP, TB), TB, 0, stream>>>(W3, W3p);
  gemm_wmma_blk<HIDD, CLSP><<<GEMM_BLOCKS, gemm_blk, 0, stream>>>(h2, W3p, xw);
  scores_k<1, CLS, CLSP><<<blks((long long)NN, TB), TB, 0, stream>>>(xw, as3, ad3, esrc, edst);
  fill_kernel<<<blks((long long)NN, TB), TB, 0, stream>>>(mseg, -INFINITY, (long long)NN);
  fill_kernel<<<blks((long long)NN, TB), TB, 0, stream>>>(den, 0.f, (long long)NN);
  fill_kernel<<<blks((long long)NN * CLS, TB), TB, 0, stream>>>(agg, 0.f, (long long)NN * CLS);
  edge_max1_k<<<blks(ET, TB), TB, 0, stream>>>(src, dst, esrc, edst, ebuf, mseg);
  edge_exp1_k<<<blks(ET, TB), TB, 0, stream>>>(dst, ebuf, mseg, den);
  edge_agg4_k<1, CLS, CLSP, CLS><<<blks((long long)ET * (CLS / 4), TB), TB, 0, stream>>>(
      src, dst, ebuf, den, xw, agg);
  out_bias_k<<<blks((long long)NN * CLS, TB), TB, 0, stream>>>(agg, b3, out);
}